// ElasticDeform_73882027426370
// MI455X (gfx1250) — compile-verified
//
#include <hip/hip_runtime.h>
#include <math.h>

// ---------------------------------------------------------------------------
// Elastic deformation, MI455X (gfx1250, wave32).
//   pass 1: vertical 41-tap Gaussian blur of (2*noise-1) via V_WMMA_F32_16X16X4_F32
//   pass 2: horizontal blur via WMMA + fused bilinear sampling of the image.
//
// Conv-as-banded-GEMM: a 16x16 output tile is Kband(16x56) x In(56x16) (or
// data x Kband for the horizontal pass), 14 chained 16x16x4 f32 WMMAs.
// Each wave processes a 4-tile strip along the conv axis: the 104-element
// input window (26 K-chunks) is loaded ONCE and shared by 4 accumulators,
// cutting halo read amplification from 3.5x to 1.625x.  The banded kernel
// fragment for tile i at chunk t is a[t-4i] -- the same 14 constant
// fragments serve every tile (k[4t+kk-16i-m] == k[4(t-4i)+kk-m]).
// Boundary handling is branchless (mask-multiply, 0*NaN!=0 blocks
// re-branching); interior strips take a wave-uniform raw-load fast path.
// ---------------------------------------------------------------------------

#define HH 4096
#define WW 4096
#define RADIUS 20
#define KLEN 41            // 2*RADIUS + 1
#define SIGMA 5.0f
#define ALPHA 20.0f
#define TW (WW / 16)       // 256 tile columns
#define THH (HH / 16)      // 256 tile rows
#define WAVES_PER_BLOCK 8
#define BLOCK 256
#define NFRAG 14           // K-chunks per 16-output band (ceil(56/4))
#define STRIP 4            // tiles per strip along the conv axis
#define NCHUNK 26          // K-chunks per strip (ceil((64+40)/4))

typedef float v2f __attribute__((ext_vector_type(2)));
typedef float v8f __attribute__((ext_vector_type(8)));

// Banded Gaussian matrix entry: k[d] if 0 <= d < 41 else 0.
__device__ __forceinline__ float bandk(const float* sk, int d) {
    return (d >= 0 && d < KLEN) ? sk[d] : 0.0f;
}

// Build the normalized Gaussian taps in LDS (once per block, trivially cheap).
__device__ __forceinline__ void fill_kernel_lds(float* sk) {
    const int t = threadIdx.x;
    if (t < KLEN) {
        float sum = 0.0f;
        for (int i = 0; i < KLEN; ++i) {
            float x = (float)(i - RADIUS) / SIGMA;
            sum += __expf(-0.5f * x * x);
        }
        float x = (float)(t - RADIUS) / SIGMA;
        sk[t] = __expf(-0.5f * x * x) / sum;
    }
    __syncthreads();
}

// ---------------------------------------------------------------------------
// Pass 1: vertical blur of g = 2*noise - 1 (zero padded in rows).
// Strip = 64 output rows x 16 cols.  Out_i = Kband_i x In(104x16).
// ---------------------------------------------------------------------------
__global__ void __launch_bounds__(BLOCK)
vblur_kernel(const float* __restrict__ nx, const float* __restrict__ ny,
             float* __restrict__ tx, float* __restrict__ ty) {
    __shared__ float sk[KLEN];
    fill_kernel_lds(sk);

    const int lane = threadIdx.x & 31;
    const int wave = threadIdx.x >> 5;
    const int g    = blockIdx.x * WAVES_PER_BLOCK + wave;   // strip id
    const int r0   = (g / TW) * (16 * STRIP);               // strip top row
    const int c0   = (g % TW) * 16;

    const int half = lane >> 4;   // K sub-offset selector: {0,1} vs {2,3}
    const int sub  = lane & 15;   // A: M index. B: N index.

    // Constant banded-kernel A fragments: a[s] = k[4s+kk - m], m = sub.
    v2f a[NFRAG];
#pragma unroll
    for (int s = 0; s < NFRAG; ++s) {
        const int j0 = 4 * s + 2 * half;
        a[s].x = bandk(sk, j0     - sub);
        a[s].y = bandk(sk, j0 + 1 - sub);
    }

    const int col = c0 + sub;
    // Rows touched: [r0-RADIUS, r0+16*STRIP+RADIUS-1].
    const bool interior = (r0 >= RADIUS) && (r0 + 16 * STRIP + RADIUS <= HH);

    const float* srcs[2] = {nx, ny};
    float*       dsts[2] = {tx, ty};

#pragma unroll
    for (int f = 0; f < 2; ++f) {
        const float* __restrict__ src = srcs[f];
        v8f c[STRIP] = {};
        if (interior) {
            const float* __restrict__ p =
                src + (size_t)(r0 - RADIUS + 2 * half) * WW + col;
#pragma unroll
            for (int t = 0; t < NCHUNK; ++t) {
                v2f b;
                b.x = fmaf(2.0f, p[(size_t)(4 * t)     * WW], -1.0f);
                b.y = fmaf(2.0f, p[(size_t)(4 * t + 1) * WW], -1.0f);
#pragma unroll
                for (int i = 0; i < STRIP; ++i) {
                    const int s = t - 4 * i;          // constant-folded guard
                    if (s >= 0 && s < NFRAG)
                        c[i] = __builtin_amdgcn_wmma_f32_16x16x4_f32(
                                   false, a[s], false, b, (short)0, c[i],
                                   false, false);
                }
            }
        } else {
#pragma unroll
            for (int t = 0; t < NCHUNK; ++t) {
                const int row = r0 - RADIUS + 4 * t + 2 * half;
                v2f b;
                {
                    const int   rcl = min(max(row, 0), HH - 1);
                    const float m   = (row >= 0 && row < HH) ? 1.0f : 0.0f;
                    b.x = m * fmaf(2.0f, src[(size_t)rcl * WW + col], -1.0f);
                }
                {
                    const int   r1  = row + 1;
                    const int   rcl = min(max(r1, 0), HH - 1);
                    const float m   = (r1 >= 0 && r1 < HH) ? 1.0f : 0.0f;
                    b.y = m * fmaf(2.0f, src[(size_t)rcl * WW + col], -1.0f);
                }
#pragma unroll
                for (int i = 0; i < STRIP; ++i) {
                    const int s = t - 4 * i;
                    if (s >= 0 && s < NFRAG)
                        c[i] = __builtin_amdgcn_wmma_f32_16x16x4_f32(
                                   false, a[s], false, b, (short)0, c[i],
                                   false, false);
                }
            }
        }
        float* __restrict__ dst = dsts[f];
#pragma unroll
        for (int i = 0; i < STRIP; ++i) {
#pragma unroll
            for (int v = 0; v < 8; ++v) {
                const int row = r0 + 16 * i + v + 8 * half;   // C/D layout
                dst[(size_t)row * WW + c0 + sub] = c[i][v];
            }
        }
    }
}

// Bilinear gather: value at (R,Ci), zero outside, branchless.
__device__ __forceinline__ float gatherpx(const float* __restrict__ img, int R, int Ci) {
    const int   rc = min(max(R, 0), HH - 1);
    const int   cc = min(max(Ci, 0), WW - 1);
    const float m  = (R >= 0 && R < HH && Ci >= 0 && Ci < WW) ? 1.0f : 0.0f;
    return m * img[(size_t)rc * WW + cc];
}

// ---------------------------------------------------------------------------
// Pass 2: horizontal blur (zero padded in cols) of the vertically blurred
// temps -> dx/dy in WMMA accumulators, then fused bilinear sampling.
// Strip = 16 rows x 64 output cols.  Out_i = In(16x104) x Kband_i.
// ---------------------------------------------------------------------------
__global__ void __launch_bounds__(BLOCK)
hblur_sample_kernel(const float* __restrict__ img,
                    const float* __restrict__ tx, const float* __restrict__ ty,
                    float* __restrict__ out) {
    __shared__ float sk[KLEN];
    fill_kernel_lds(sk);

    const int lane = threadIdx.x & 31;
    const int wave = threadIdx.x >> 5;
    const int g    = blockIdx.x * WAVES_PER_BLOCK + wave;        // strip id
    const int r0   = (g / (WW / (16 * STRIP))) * 16;             // tile row
    const int c0   = (g % (WW / (16 * STRIP))) * (16 * STRIP);   // strip left

    const int half = lane >> 4;
    const int sub  = lane & 15;

    // Warm the gather footprint while the WMMA conv chain runs:
    // 16 rows x 64 cols = 16 x 2 cachelines = exactly 32 lanes.
    __builtin_prefetch(img + (size_t)(r0 + sub) * WW + c0 + 32 * half, 0, 3);

    // Constant banded-kernel B fragments: bk[s] = k[4s+kk - n], n = sub.
    v2f bk[NFRAG];
#pragma unroll
    for (int s = 0; s < NFRAG; ++s) {
        const int j0 = 4 * s + 2 * half;
        bk[s].x = bandk(sk, j0     - sub);
        bk[s].y = bandk(sk, j0 + 1 - sub);
    }

    const int arow = r0 + sub;            // A-fragment row (M = sub)
    // Columns touched: [c0-RADIUS, c0+16*STRIP+RADIUS-1].
    const bool interior = (c0 >= RADIUS) && (c0 + 16 * STRIP + RADIUS <= WW);

    const float* tsrc[2] = {tx, ty};
    v8f acc[2][STRIP];

#pragma unroll
    for (int f = 0; f < 2; ++f) {
        const float* __restrict__ src = tsrc[f];
        v8f c[STRIP] = {};
        if (interior) {
            // The two A elements are adjacent columns, 8B aligned
            // -> single global_load_b64 per chunk.
            const float* __restrict__ p =
                src + (size_t)arow * WW + (c0 - RADIUS + 2 * half);
#pragma unroll
            for (int t = 0; t < NCHUNK; ++t) {
                const v2f av = *(const v2f*)(p + 4 * t);
#pragma unroll
                for (int i = 0; i < STRIP; ++i) {
                    const int s = t - 4 * i;
                    if (s >= 0 && s < NFRAG)
                        c[i] = __builtin_amdgcn_wmma_f32_16x16x4_f32(
                                   false, av, false, bk[s], (short)0, c[i],
                                   false, false);
                }
            }
        } else {
#pragma unroll
            for (int t = 0; t < NCHUNK; ++t) {
                const int jc = c0 - RADIUS + 4 * t + 2 * half;
                v2f av;
                {
                    const int   ccl = min(max(jc, 0), WW - 1);
                    const float m   = (jc >= 0 && jc < WW) ? 1.0f : 0.0f;
                    av.x = m * src[(size_t)arow * WW + ccl];
                }
                {
                    const int   j1  = jc + 1;
                    const int   ccl = min(max(j1, 0), WW - 1);
                    const float m   = (j1 >= 0 && j1 < WW) ? 1.0f : 0.0f;
                    av.y = m * src[(size_t)arow * WW + ccl];
                }
#pragma unroll
                for (int i = 0; i < STRIP; ++i) {
                    const int s = t - 4 * i;
                    if (s >= 0 && s < NFRAG)
                        c[i] = __builtin_amdgcn_wmma_f32_16x16x4_f32(
                                   false, av, false, bk[s], (short)0, c[i],
                                   false, false);
                }
            }
        }
#pragma unroll
        for (int i = 0; i < STRIP; ++i) acc[f][i] = c[i];
    }

    // Fused bilinear sampling.  C/D layout: row r0+v+8*half, col c0+16i+sub.
    // bilinear(img/255)*255 == bilinear(img) under zero padding (linearity).
#pragma unroll
    for (int i = 0; i < STRIP; ++i) {
#pragma unroll
        for (int v = 0; v < 8; ++v) {
            const int r  = r0 + v + 8 * half;
            const int cc = c0 + 16 * i + sub;
            const float rr  = (float)r  + ALPHA * acc[0][i][v]; // dx -> rows
            const float ccf = (float)cc + ALPHA * acc[1][i][v]; // dy -> cols

            const float rf = floorf(rr), cf = floorf(ccf);
            const float wr = rr - rf,    wc = ccf - cf;
            const int ri = (int)rf, ci = (int)cf;

            const float v00 = gatherpx(img, ri,     ci);
            const float v01 = gatherpx(img, ri,     ci + 1);
            const float v10 = gatherpx(img, ri + 1, ci);
            const float v11 = gatherpx(img, ri + 1, ci + 1);

            const float res = v00 * (1.0f - wr) * (1.0f - wc)
                            + v01 * (1.0f - wr) * wc
                            + v10 * wr * (1.0f - wc)
                            + v11 * wr * wc;
            out[(size_t)r * WW + cc] = res;
        }
    }
}

extern "C" void kernel_launch(void* const* d_in, const int* in_sizes, int n_in,
                              void* d_out, int out_size, void* d_ws, size_t ws_size,
                              hipStream_t stream) {
    (void)in_sizes; (void)n_in; (void)out_size; (void)ws_size;
    const float* img = (const float*)d_in[0];
    const float* nx  = (const float*)d_in[1];
    const float* ny  = (const float*)d_in[2];
    float* out = (float*)d_out;
    float* tX  = (float*)d_ws;                       // 64 MB vertical-blur temp X
    float* tY  = tX + (size_t)HH * WW;               // 64 MB vertical-blur temp Y

    // Pass 1: (HH/64) x (WW/16) = 16384 strips, 8 waves/block.
    const int nstrip1 = (HH / (16 * STRIP)) * TW;
    vblur_kernel<<<dim3(nstrip1 / WAVES_PER_BLOCK), BLOCK, 0, stream>>>(nx, ny, tX, tY);

    // Pass 2: (HH/16) x (WW/64) = 16384 strips, 8 waves/block.
    const int nstrip2 = THH * (WW / (16 * STRIP));
    hblur_sample_kernel<<<dim3(nstrip2 / WAVES_PER_BLOCK), BLOCK, 0, stream>>>(img, tX, tY, out);
}